// GraphEnhancedTemporalModel_69630009802983
// MI455X (gfx1250) — compile-verified
//
#include <hip/hip_runtime.h>

#define NUM_NODES 100000
#define NUM_EDGES 640000
#define D_IN 128
#define D_OUT 128
#define NUM_REL 5
#define WT_STRIDE 132   // 132 % 64 == 4  -> conflict-free b64 B-fragment LDS reads
#define BLOCKS_X 160    // per-relation blocks; waves grid-stride over edge tiles

typedef __attribute__((ext_vector_type(2))) float v2f;
typedef __attribute__((ext_vector_type(8))) float v8f;

// ---------------- init: out = bias broadcast, zero relation counters --------
__global__ void init_out_kernel(float* __restrict__ out,
                                const float* __restrict__ bias,
                                int* __restrict__ counts) {
    int i = blockIdx.x * blockDim.x + threadIdx.x;
    if (i < NUM_NODES * D_OUT) out[i] = bias[i & (D_OUT - 1)];
    if (i < NUM_REL) counts[i] = 0;
}

// ---------------- histogram edges by relation -------------------------------
__global__ void count_rel_kernel(const int* __restrict__ edge_type,
                                 int* __restrict__ counts) {
    int e = blockIdx.x * blockDim.x + threadIdx.x;
    if (e >= NUM_EDGES) return;
    int r = edge_type[e];
    r = r < 0 ? 0 : (r > NUM_REL - 1 ? NUM_REL - 1 : r);
    atomicAdd(&counts[r], 1);
}

// ---------------- tiny prefix sum over 5 buckets ----------------------------
__global__ void offsets_kernel(const int* __restrict__ counts,
                               int* __restrict__ offsets,
                               int* __restrict__ cursors) {
    if (threadIdx.x == 0 && blockIdx.x == 0) {
        int acc = 0;
        for (int r = 0; r < NUM_REL; ++r) {
            offsets[r] = acc;
            cursors[r] = acc;
            acc += counts[r];
        }
    }
}

// ---------------- bucket edge ids by relation -------------------------------
__global__ void scatter_perm_kernel(const int* __restrict__ edge_type,
                                    int* __restrict__ cursors,
                                    int* __restrict__ perm) {
    int e = blockIdx.x * blockDim.x + threadIdx.x;
    if (e >= NUM_EDGES) return;
    int r = edge_type[e];
    r = r < 0 ? 0 : (r > NUM_REL - 1 ? NUM_REL - 1 : r);
    int pos = atomicAdd(&cursors[r], 1);
    perm[pos] = e;
}

// ---------------- fused gather -> WMMA f32 GEMM -> scaled scatter-add -------
// grid = (BLOCKS_X, NUM_REL), block = 256 (8 waves). Each wave grid-strides
// over 32-edge tiles of its relation: two 16x128 A fragments share every
// B fragment (halves LDS traffic per WMMA), 16 WMMAs per K-step of 4.
__launch_bounds__(256)
__global__ void rgcn_wmma_kernel(const float* __restrict__ x,
                                 const int* __restrict__ edge_index, // [2, E] flat
                                 const float* __restrict__ edge_attr,
                                 const float* __restrict__ weight,   // [R, 128, 128]
                                 const int* __restrict__ counts,
                                 const int* __restrict__ offsets,
                                 const int* __restrict__ perm,
                                 float* __restrict__ out) {
    __shared__ float Wt[D_OUT * WT_STRIDE];   // Wt[n*S + k] = W[k][n] (transposed)

    const int r = blockIdx.y;
    const int cnt = counts[r];
    if (cnt == 0) return;
    const int off = offsets[r];

    // cooperative transposed load of W[r] into LDS (coalesced float4 reads)
    const float* Wg = weight + (size_t)r * D_IN * D_OUT;
    for (int i = threadIdx.x; i < (D_IN * D_OUT) / 4; i += 256) {
        const float4 w4 = ((const float4*)Wg)[i];
        const int idx = i * 4;
        const int k = idx >> 7;     // d_in row
        const int n = idx & 127;    // d_out col
        Wt[(n + 0) * WT_STRIDE + k] = w4.x;
        Wt[(n + 1) * WT_STRIDE + k] = w4.y;
        Wt[(n + 2) * WT_STRIDE + k] = w4.z;
        Wt[(n + 3) * WT_STRIDE + k] = w4.w;
    }
    __syncthreads();

    const int lane = threadIdx.x & 31;
    const int wave = threadIdx.x >> 5;
    const int nl   = lane & 15;
    const int half = lane >> 4;     // 0: K-pair {0,1}; 1: K-pair {2,3}
    const int sub  = half * 2;

    const int ntiles  = (cnt + 31) >> 5;       // 32 edges per tile
    const int wstride = gridDim.x * 8;

    for (int tile = blockIdx.x * 8 + wave; tile < ntiles; tile += wstride) {
        // every lane owns one of the tile's 32 edges; pad with harmless zeros
        int src = 0, dst = 0;
        float attr = 0.0f;
        const int eo = tile * 32 + lane;
        if (eo < cnt) {
            const int e = perm[off + eo];
            src  = edge_index[e];              // row 0: src
            dst  = edge_index[NUM_EDGES + e];  // row 1: dst
            attr = edge_attr[e];
        }

        // A-fragment rows: fragment0 <- edges 0..15, fragment1 <- edges 16..31
        const int src0 = __shfl(src, nl, 32);
        const int src1 = __shfl(src, nl + 16, 32);
        const float* xrow0 = x + (size_t)src0 * D_IN;
        const float* xrow1 = x + (size_t)src1 * D_IN;

        v8f acc0[8] = {};
        v8f acc1[8] = {};

        for (int k = 0; k < D_IN; k += 4) {
            const v2f a0 = *(const v2f*)(xrow0 + k + sub);
            const v2f a1 = *(const v2f*)(xrow1 + k + sub);
#pragma unroll
            for (int t = 0; t < 8; ++t) {
                // B: 4x16 f32 from transposed LDS, shared by both A fragments
                const v2f b = *(const v2f*)(&Wt[(t * 16 + nl) * WT_STRIDE + k + sub]);
                acc0[t] = __builtin_amdgcn_wmma_f32_16x16x4_f32(
                    false, a0, false, b, (short)0, acc0[t], false, false);
                acc1[t] = __builtin_amdgcn_wmma_f32_16x16x4_f32(
                    false, a1, false, b, (short)0, acc1[t], false, false);
            }
        }

        // C/D layout: VGPR j -> M = j + 8*half (edge index in tile), N = nl + 16*t
#pragma unroll
        for (int j = 0; j < 8; ++j) {
            const int   M0  = j + half * 8;
            const int   M1  = M0 + 16;
            const float at0 = __shfl(attr, M0, 32);
            const int   d0  = __shfl(dst,  M0, 32);
            const float at1 = __shfl(attr, M1, 32);
            const int   d1  = __shfl(dst,  M1, 32);
            float* o0 = out + (size_t)d0 * D_OUT;
            float* o1 = out + (size_t)d1 * D_OUT;
#pragma unroll
            for (int t = 0; t < 8; ++t) {
                atomicAdd(&o0[t * 16 + nl], acc0[t][j] * at0);
                atomicAdd(&o1[t * 16 + nl], acc1[t][j] * at1);
            }
        }
    }
}

// ---------------------------------------------------------------------------
extern "C" void kernel_launch(void* const* d_in, const int* in_sizes, int n_in,
                              void* d_out, int out_size, void* d_ws, size_t ws_size,
                              hipStream_t stream) {
    const float* x          = (const float*)d_in[0];
    const int*   edge_index = (const int*)d_in[1];
    const int*   edge_type  = (const int*)d_in[2];
    const float* edge_attr  = (const float*)d_in[3];
    const float* weight     = (const float*)d_in[4];
    const float* bias       = (const float*)d_in[5];
    float*       out        = (float*)d_out;

    int* ws_i    = (int*)d_ws;
    int* counts  = ws_i;        // [5]
    int* offsets = ws_i + 8;    // [5]
    int* cursors = ws_i + 16;   // [5]
    int* perm    = ws_i + 32;   // [NUM_EDGES]  (~2.56 MB workspace total)

    {
        const int n = NUM_NODES * D_OUT;
        init_out_kernel<<<(n + 255) / 256, 256, 0, stream>>>(out, bias, counts);
    }
    count_rel_kernel<<<(NUM_EDGES + 255) / 256, 256, 0, stream>>>(edge_type, counts);
    offsets_kernel<<<1, 32, 0, stream>>>(counts, offsets, cursors);
    scatter_perm_kernel<<<(NUM_EDGES + 255) / 256, 256, 0, stream>>>(edge_type, cursors, perm);

    dim3 grid(BLOCKS_X, NUM_REL, 1);   // waves grid-stride over 32-edge tiles
    rgcn_wmma_kernel<<<grid, 256, 0, stream>>>(x, edge_index, edge_attr, weight,
                                               counts, offsets, perm, out);
}